// HyperAttentionEmbedding_24043226923189
// MI455X (gfx1250) — compile-verified
//
#include <hip/hip_runtime.h>
#include <hip/hip_bf16.h>

// HyperAttentionEmbedding for MI455X (gfx1250).
// fp32 matrix ops via V_WMMA_F32_16X16X4_F32; z tiles fetched by the
// Tensor Data Mover (TENSOR_LOAD_TO_LDS, TENSORcnt) to offload the
// roofline-dominant 67MB z stream from the vector pipe.
// B=1, N=1024, H=8, C=64, NBR=32, R=4

#define C_DIM 64
#define NBR_N 32
#define HEADS 8
#define NPTS 1024
#define BH    8          // B*H
#define QT    2          // queries per block in kernel 1
#define EPS   1e-6f

typedef __attribute__((ext_vector_type(2))) float    v2f;
typedef __attribute__((ext_vector_type(8))) float    v8f;
typedef __attribute__((ext_vector_type(4))) unsigned v4u;
typedef __attribute__((ext_vector_type(8))) int      v8i;
typedef __attribute__((ext_vector_type(4))) int      v4i;

static __device__ inline v8f wmma_f32_4(v2f a, v2f b, v8f c) {
  // D = A(16x4) * B(4x16) + C(16x16), fp32
  return __builtin_amdgcn_wmma_f32_16x16x4_f32(false, a, false, b, (short)0, c,
                                               false, false);
}

static __device__ inline v8f vzero8() {
  v8f z;
#pragma unroll
  for (int i = 0; i < 8; ++i) z[i] = 0.0f;
  return z;
}

// ---------------------------------------------------------------------------
// Kernel 1: TDM-load z tile, gather x_j, (x_j @ Ws) and (z @ Wz) via WMMA,
//           masked neighbor reduction + LayerNorm -> x_conv (workspace).
// block: 64 threads (2 waves); each wave owns one query point of one head.
// ---------------------------------------------------------------------------
__global__ __launch_bounds__(64) void conv_ln_kernel(
    const float* __restrict__ q_i, const float* __restrict__ k_i,
    const float* __restrict__ m_j, const float* __restrict__ m_ij,
    const float* __restrict__ z_ij, const int* __restrict__ nidx,
    const float* __restrict__ Ws_q, const float* __restrict__ Wz_q,
    const float* __restrict__ g_q,  const float* __restrict__ b_q,
    const float* __restrict__ Ws_k, const float* __restrict__ Wz_k,
    const float* __restrict__ g_k,  const float* __restrict__ b_k,
    float* __restrict__ xconv)
{
  const int branch = blockIdx.y;
  const float* __restrict__ x     = branch ? k_i  : q_i;
  const float* __restrict__ Ws    = branch ? Ws_k : Ws_q;
  const float* __restrict__ Wz    = branch ? Wz_k : Wz_q;
  const float* __restrict__ gamma = branch ? g_k  : g_q;
  const float* __restrict__ beta  = branch ? b_k  : b_q;

  const int blocksPerHead = NPTS / QT;                  // 512
  const int bh    = blockIdx.x / blocksPerHead;
  const int nbase = (blockIdx.x % blocksPerHead) * QT;

  // Single shared array so descriptor LDS byte offsets are well defined
  // (offset 0 == start of smem).
  __shared__ float smem[4096 + 4096 + QT * C_DIM + QT * NBR_N];
  float* sXj  = smem;            // [QT*NBR*64] 16 KB; later holds A = Xj@Ws
  float* sZt  = smem + 4096;     // [QT*NBR*64] 16 KB; later holds B = Z@Wz
  float* sXi  = smem + 8192;     // [QT*64]
  float* sMsk = smem + 8192 + QT * C_DIM;  // [QT*32]

  const int t = threadIdx.x;                            // 0..63
  const int wv = t >> 5, lane = t & 31;

  // ---- TDM: per-wave async DMA of this query's z tile into sZt ------------
  // Tile: 32 rows x 64 fp32, row stride = H*C = 512 elements.
  {
    const int n_wv = nbase + wv;   // wave-uniform
    const unsigned long long gaddr = (unsigned long long)(const void*)z_ij +
        ((unsigned long long)((size_t)n_wv * NBR_N * HEADS + bh) * C_DIM) * 4ull;
    unsigned ga_lo = (unsigned)__builtin_amdgcn_readfirstlane((int)(unsigned)gaddr);
    unsigned ga_hi = (unsigned)__builtin_amdgcn_readfirstlane((int)(unsigned)(gaddr >> 32));
    unsigned ldsb  = (unsigned)__builtin_amdgcn_readfirstlane(
                         (int)(16384u + (unsigned)wv * 8192u));
    // D# group0: count=1 | lds_addr | global_addr[56:0] | type=2
    v4u g0 = { 1u, ldsb, ga_lo, (ga_hi & 0x01FFFFFFu) | (2u << 30) };
    // D# group1: data_size=4B; tensor_dim0=64; tensor_dim1=32;
    //            tile_dim0=64; tile_dim1=32; dim0_stride=512
    v8i g1 = { 0x00020000,          // wg_mask=0, data_size=2 (4B)
               0x00400000,          // ab_addr=0, tensor_dim0 lo16 = 64
               0x00200000,          // tensor_dim0 hi16=0, tensor_dim1 lo16 = 32
               0x00400000,          // tensor_dim1 hi16=0, tile_dim0 = 64
               32,                  // tile_dim1 = 32, tile_dim2 = 0
               512,                 // tensor_dim0_stride lo32 = 512
               0, 0 };
    v4i gz4  = { 0, 0, 0, 0 };
    v8i gz8  = { 0, 0, 0, 0, 0, 0, 0, 0 };
    // 6-arg toolchain form: (g0, g1, g2, g3, g4, cpol)
    __builtin_amdgcn_tensor_load_to_lds(g0, g1, gz4, gz4, gz8, 0);
  }

  // ---- gather x_j: row t = (query t/32, neighbor t%32) ---------------------
  {
    const int q = t / NBR_N, j = t % NBR_N;
    const int n = nbase + q;
    const int gi = nidx[n * NBR_N + j];
    const float4* s1 = (const float4*)(x + ((size_t)gi * HEADS + bh) * C_DIM);
    float4* d1 = (float4*)(sXj + t * C_DIM);
#pragma unroll
    for (int i = 0; i < C_DIM / 4; ++i) d1[i] = s1[i];
    sMsk[t] = m_ij[n * NBR_N + j] * m_j[n * NBR_N + j];
  }
  // ---- x_i rows (QT*64 floats) --------------------------------------------
#pragma unroll
  for (int e = t; e < QT * C_DIM; e += 64) {
    const int q = e / C_DIM, c = e % C_DIM;
    sXi[e] = x[(((size_t)(nbase + q)) * HEADS + bh) * C_DIM + c];
  }
  __syncthreads();

  const int lm = lane & 15, hi = lane >> 4, kb = hi * 2;

  // ---- two GEMMs, results overwrite sXj / sZt in place ---------------------
  // wave wv owns rows [32*wv, 32*wv+32) (== its query's 32 neighbors).
  for (int g = 0; g < 2; ++g) {
    if (g == 1) {
      // first use of the TDM-loaded z tile (wave-private region)
      __builtin_amdgcn_s_wait_tensorcnt(0);
    }
    float* __restrict__ src = g ? sZt : sXj;
    const float* __restrict__ W = g ? Wz : Ws;
#pragma unroll
    for (int mt = 0; mt < 2; ++mt) {
      const int mrow = wv * 32 + mt * 16;
      v8f acc[4];
#pragma unroll
      for (int nt = 0; nt < 4; ++nt) acc[nt] = vzero8();
      for (int ks = 0; ks < 16; ++ks) {
        const int k0 = ks * 4;
        v2f a;
        a.x = src[(mrow + lm) * C_DIM + k0 + kb];
        a.y = src[(mrow + lm) * C_DIM + k0 + kb + 1];
#pragma unroll
        for (int nt = 0; nt < 4; ++nt) {
          v2f b;
          b.x = W[(k0 + kb)     * C_DIM + nt * 16 + lm];
          b.y = W[(k0 + kb + 1) * C_DIM + nt * 16 + lm];
          acc[nt] = wmma_f32_4(a, b, acc[nt]);
        }
      }
      // write D tile back over consumed rows (own-wave rows only)
#pragma unroll
      for (int nt = 0; nt < 4; ++nt)
#pragma unroll
        for (int v = 0; v < 8; ++v)
          src[(mrow + v + 8 * hi) * C_DIM + nt * 16 + lm] = acc[nt][v];
    }
  }

  // ---- masked neighbor reduction + LayerNorm (wave wv -> query wv) ---------
  const float xi0 = sXi[wv * C_DIM + lane];
  const float xi1 = sXi[wv * C_DIM + lane + 32];
  float a0 = 0.f, a1 = 0.f;
  for (int j = 0; j < NBR_N; ++j) {
    const int row = wv * NBR_N + j;
    const float mj = sMsk[wv * NBR_N + j];
    const float A0 = sXj[row * C_DIM + lane];
    const float A1 = sXj[row * C_DIM + lane + 32];
    const float B0 = sZt[row * C_DIM + lane];
    const float B1 = sZt[row * C_DIM + lane + 32];
    a0 += mj * (A0 - xi0) * B0;
    a1 += mj * (A1 - xi1) * B1;
  }
  float s = a0 + a1, ss = a0 * a0 + a1 * a1;
#pragma unroll
  for (int off = 16; off > 0; off >>= 1) {
    s  += __shfl_xor(s,  off, 32);
    ss += __shfl_xor(ss, off, 32);
  }
  const float mu  = s * (1.f / 64.f);
  const float var = ss * (1.f / 64.f) - mu * mu;
  const float inv = rsqrtf(var + EPS);
  const float o0 = (a0 - mu) * inv * gamma[lane]      + beta[lane];
  const float o1 = (a1 - mu) * inv * gamma[lane + 32] + beta[lane + 32];

  const size_t gr = (size_t)bh * NPTS + (nbase + wv);
  float* op = xconv + (size_t)branch * BH * NPTS * C_DIM + gr * C_DIM;
  op[lane]      = o0;
  op[lane + 32] = o1;
}

// ---------------------------------------------------------------------------
// Kernel 2: L = x_conv @ Wl, R = x_conv @ Wr (WMMA), banded dw = L R^T (|i-k|<=4),
//           out = x + dw @ x, permuted to (B,N,H,C), q||k concatenated.
// block: 128 threads (4 waves) handles 16 consecutive (bh,n) rows.
// ---------------------------------------------------------------------------
__global__ __launch_bounds__(128) void hak_band_kernel(
    const float* __restrict__ q_i, const float* __restrict__ k_i,
    const float* __restrict__ Wl_q, const float* __restrict__ Wr_q,
    const float* __restrict__ Wl_k, const float* __restrict__ Wr_k,
    const float* __restrict__ xconv, float* __restrict__ out)
{
  const int branch = blockIdx.y;
  const float* __restrict__ x  = branch ? k_i  : q_i;
  const float* __restrict__ Wl = branch ? Wl_k : Wl_q;
  const float* __restrict__ Wr = branch ? Wr_k : Wr_q;
  const int rbase = blockIdx.x * 16;

  __shared__ float sXc[16 * C_DIM];     // conv rows        (4 KB)
  __shared__ float sXo[16 * C_DIM];     // original x rows  (4 KB)
  __shared__ float sL[16 * 256];        // 16 KB
  __shared__ float sR[16 * 256];        // 16 KB

  const int t = threadIdx.x;            // 0..127
  // ---- load 16 rows of x_conv and original x (8 threads per row) -----------
  {
    const int ri = t >> 3, seg = t & 7;
    const int gr = rbase + ri;
    const int bh = gr >> 10, n = gr & 1023;
    const float4* xc = (const float4*)(xconv +
        (size_t)branch * BH * NPTS * C_DIM + (size_t)gr * C_DIM);
    const float4* xo = (const float4*)(x + ((size_t)n * HEADS + bh) * C_DIM);
    ((float4*)(sXc + ri * C_DIM))[seg * 2]     = xc[seg * 2];
    ((float4*)(sXc + ri * C_DIM))[seg * 2 + 1] = xc[seg * 2 + 1];
    ((float4*)(sXo + ri * C_DIM))[seg * 2]     = xo[seg * 2];
    ((float4*)(sXo + ri * C_DIM))[seg * 2 + 1] = xo[seg * 2 + 1];
  }
  __syncthreads();

  const int wv = t >> 5, lane = t & 31;
  const int lm = lane & 15, hi = lane >> 4, kb = hi * 2;

  // ---- GEMMs: wave wv computes cols [64*wv, 64*wv+64) of L and R -----------
  for (int g = 0; g < 2; ++g) {
    const float* __restrict__ W = g ? Wr : Wl;
    float* __restrict__ dst = g ? sR : sL;
    v8f acc[4];
#pragma unroll
    for (int nt = 0; nt < 4; ++nt) acc[nt] = vzero8();
    for (int ks = 0; ks < 16; ++ks) {
      const int k0 = ks * 4;
      v2f a;
      a.x = sXc[lm * C_DIM + k0 + kb];
      a.y = sXc[lm * C_DIM + k0 + kb + 1];
#pragma unroll
      for (int nt = 0; nt < 4; ++nt) {
        const int col = wv * 64 + nt * 16 + lm;
        v2f b;
        b.x = W[(k0 + kb)     * 256 + col];
        b.y = W[(k0 + kb + 1) * 256 + col];
        acc[nt] = wmma_f32_4(a, b, acc[nt]);
      }
    }
#pragma unroll
    for (int nt = 0; nt < 4; ++nt)
#pragma unroll
      for (int v = 0; v < 8; ++v)
        dst[(v + 8 * hi) * 256 + wv * 64 + nt * 16 + lm] = acc[nt][v];
  }
  __syncthreads();

  // ---- band update: thread -> (row ri, 8 output channels) ------------------
  {
    const int ri = t >> 3;
    const int cb = (t & 7) * 8;
    const int gr = rbase + ri;
    const int bh = gr >> 10, n = gr & 1023;
    float* op = out + (size_t)branch * NPTS * HEADS * C_DIM +
                ((size_t)n * HEADS + bh) * C_DIM;
#pragma unroll
    for (int cc = 0; cc < 8; ++cc) {
      const int c = cb + cc;
      float acc = sXo[ri * C_DIM + c];
      const int klo = (c - 4 < 0) ? 0 : c - 4;
      const int khi = (c + 4 > 63) ? 63 : c + 4;
      for (int k = klo; k <= khi; ++k) {
        float dw = 0.f;
#pragma unroll
        for (int r = 0; r < 4; ++r)
          dw += sL[ri * 256 + c * 4 + r] * sR[ri * 256 + k * 4 + r];
        acc += dw * sXo[ri * C_DIM + k];
      }
      op[c] = acc;
    }
  }
}

// ---------------------------------------------------------------------------
extern "C" void kernel_launch(void* const* d_in, const int* in_sizes, int n_in,
                              void* d_out, int out_size, void* d_ws, size_t ws_size,
                              hipStream_t stream) {
  const float* q_i  = (const float*)d_in[0];
  const float* k_i  = (const float*)d_in[1];
  const float* m_j  = (const float*)d_in[2];
  const float* m_ij = (const float*)d_in[3];
  const float* z_ij = (const float*)d_in[4];
  const int*   nidx = (const int*)  d_in[5];
  const float* Ws_q = (const float*)d_in[6];
  const float* Wz_q = (const float*)d_in[7];
  const float* g_q  = (const float*)d_in[8];
  const float* b_q  = (const float*)d_in[9];
  const float* Wl_q = (const float*)d_in[10];
  const float* Wr_q = (const float*)d_in[11];
  const float* Ws_k = (const float*)d_in[12];
  const float* Wz_k = (const float*)d_in[13];
  const float* g_k  = (const float*)d_in[14];
  const float* b_k  = (const float*)d_in[15];
  const float* Wl_k = (const float*)d_in[16];
  const float* Wr_k = (const float*)d_in[17];

  float* xconv = (float*)d_ws;   // 2 * 8 * 1024 * 64 floats = 4 MB

  dim3 g1(BH * NPTS / QT, 2);    // 4096 x 2, 64 threads
  conv_ln_kernel<<<g1, 64, 0, stream>>>(q_i, k_i, m_j, m_ij, z_ij, nidx,
                                        Ws_q, Wz_q, g_q, b_q,
                                        Ws_k, Wz_k, g_k, b_k, xconv);

  dim3 g2(BH * NPTS / 16, 2);    // 512 x 2, 128 threads
  hak_band_kernel<<<g2, 128, 0, stream>>>(q_i, k_i, Wl_q, Wr_q, Wl_k, Wr_k,
                                          xconv, (float*)d_out);
}